// v_LPTC_NG_7808250544863
// MI455X (gfx1250) — compile-verified
//
#include <hip/hip_runtime.h>

typedef __attribute__((ext_vector_type(2))) float v2f;
typedef __attribute__((ext_vector_type(4))) float v4f;
typedef __attribute__((ext_vector_type(8))) float v8f;

#define FIN   2
#define FOUT  2
#define NPATH 9
#define GX 16
#define GY 16
#define GZ 16
#define GT 32
#define VOL (GX * GY * GZ * GT) /* 131072 sites */

// One wave per site.  Per path p: stage this site's 4 (f,o) 4x4 complex
// matrices (512 B) into LDS with one coalesced non-temporal b128 load per
// lane, expand into the real 16x16 matrix [[Re,-Im],[Im,Re]] in the WMMA
// A-operand layout, broadcast the shifted 16-real feature vector into the
// B operand (all 16 columns equal), and accumulate with 4 chained
// v_wmma_f32_16x16x4_f32 (K = 16).
__global__ __launch_bounds__(256)
void lptc_wmma_kernel(const float* __restrict__ feat,
                      const float* __restrict__ wgt,
                      float* __restrict__ out)
{
    __shared__ __align__(16) float lds[8 * 2 * 128]; // 8 waves x dbl-buffer x 512B

    const int tid  = threadIdx.x;
    const int lane = tid & 31;
    const int wid  = tid >> 5;
    const int site = (blockIdx.x << 3) | wid;

    const int t = site & (GT - 1);
    const int z = (site >> 5) & (GZ - 1);
    const int y = (site >> 9) & (GY - 1);
    const int x = (site >> 13) & (GX - 1);

    float* wlds = &lds[wid * 256];

    // WMMA A layout: row m = lane%16; K element = j + 2*(lane/16) + 4*chunk
    const int h      = lane >> 4;
    const int m      = lane & 15;
    const int im_row = m >> 3;        // 0 = real output rows, 1 = imag rows
    const int o_     = (m >> 2) & 1;
    const int a_     = m & 3;

    // weight staging decode: lane -> (f*2+o) matrix, 16-byte chunk
    const int mfo = lane >> 3;
    const int c16 = lane & 7;

    const v4f* Wq = (const v4f*)wgt;
    const v4f* Fq = (const v4f*)feat;

    constexpr int PMU[NPATH] = {-1, 0, 0, 1, 1, 2, 2, 3, 3};
    constexpr int PNH[NPATH] = { 0, 1,-1, 1,-1, 1,-1, 1,-1};

    v8f acc = {0.f, 0.f, 0.f, 0.f, 0.f, 0.f, 0.f, 0.f};

    // software prefetch of path 0 weights (non-temporal: 604MB read-once stream)
    v4f wv4 = __builtin_nontemporal_load(
        &Wq[((size_t)(mfo * NPATH + 0) * VOL + site) * 8 + c16]);

    #pragma unroll
    for (int p = 0; p < NPATH; ++p) {
        float* buf = wlds + (p & 1) * 128;
        *(v4f*)&buf[lane * 4] = wv4;        // ds_store_b128

        if (p + 1 < NPATH) {                // prefetch next path while we compute
            wv4 = __builtin_nontemporal_load(
                &Wq[((size_t)(mfo * NPATH + (p + 1)) * VOL + site) * 8 + c16]);
        }

        // shifted source site: roll(v, -nh, mu) gathers coord+nh (mod dim)
        int cc[4] = {x, y, z, t};
        if (PMU[p] >= 0) {
            const int mu = PMU[p];
            const int d  = (mu == 3) ? GT : 16;
            cc[mu] = (cc[mu] + PNH[p] + d) & (d - 1);
        }
        const int src = ((cc[0] * GY + cc[1]) * GZ + cc[2]) * GT + cc[3];

        // wave-uniform feature gather (L2-resident; hardware-broadcast)
        // xk[k]: k<8 -> Re(f=k>>2, b=k&3), k>=8 -> Im
        float xk[16];
        #pragma unroll
        for (int f = 0; f < FIN; ++f) {
            #pragma unroll
            for (int half = 0; half < 2; ++half) {
                v4f fv = Fq[((size_t)f * VOL + src) * 2 + half];
                xk[f * 4 + half * 2 + 0]     = fv[0];
                xk[8 + f * 4 + half * 2 + 0] = fv[1];
                xk[f * 4 + half * 2 + 1]     = fv[2];
                xk[8 + f * 4 + half * 2 + 1] = fv[3];
            }
        }

        // 4 chained WMMAs: K = 4c + 2h + j sweeps the 16-wide real contraction
        #pragma unroll
        for (int c = 0; c < 4; ++c) {
            const int f_     = c & 1;   // (K&7)>>2 : input-feature block
            const int im_col = c >> 1;  // K>=8     : imaginary input columns
            v2f Aop, Bop;
            #pragma unroll
            for (int j = 0; j < 2; ++j) {
                const int b_  = 2 * h + j;
                const int off = ((f_ * 2 + o_) * 16 + (a_ * 4 + b_)) * 2
                                + (im_row ^ im_col);
                float wval = buf[off];                 // ds_load_b32, conflict-free
                if (im_col) wval = im_row ? wval : -wval;  // -Im block of [[Re,-Im],[Im,Re]]
                Aop[j] = wval;
                Bop[j] = h ? xk[4 * c + j + 2] : xk[4 * c + j];
            }
            acc = __builtin_amdgcn_wmma_f32_16x16x4_f32(
                false, Aop, false, Bop, (short)0, acc, false, false);
        }
    }

    // All 16 D-columns are identical; lane 0 holds m=0..7 (real parts),
    // lane 16 holds m=8..15 (imag parts) in acc[0..7].
    if ((lane & 15) == 0) {
        #pragma unroll
        for (int vc = 0; vc < 8; ++vc) {
            const int oo = vc >> 2;
            const int aa = vc & 3;
            out[(((size_t)oo * VOL + site) * 4 + aa) * 2 + h] = acc[vc];
        }
    }
}

extern "C" void kernel_launch(void* const* d_in, const int* in_sizes, int n_in,
                              void* d_out, int out_size, void* d_ws, size_t ws_size,
                              hipStream_t stream) {
    const float* feat = (const float*)d_in[0];  // features_in, complex64 -> float pairs
    const float* wgt  = (const float*)d_in[1];  // weights,     complex64 -> float pairs
    float* out = (float*)d_out;                 // complex64 output -> float pairs

    dim3 grid(VOL / 8);   // 8 waves (sites) per 256-thread block
    dim3 block(256);
    hipLaunchKernelGGL(lptc_wmma_kernel, grid, block, 0, stream, feat, wgt, out);
}